// GCN2_39642548142777
// MI455X (gfx1250) — compile-verified
//
#include <hip/hip_runtime.h>
#include <hip/hip_bf16.h>

typedef __attribute__((ext_vector_type(2))) float v2f;
typedef __attribute__((ext_vector_type(8))) float v8f;

#define TPB 256

// ---------------------------------------------------------------- utilities
__global__ void fill_f32(float* __restrict__ p, float v, long long n) {
    long long i = (long long)blockIdx.x * blockDim.x + threadIdx.x;
    if (i < n) p[i] = v;
}

__global__ void rsqrt_inplace(float* __restrict__ p, int n) {
    int i = blockIdx.x * blockDim.x + threadIdx.x;
    if (i < n) {
        float d = p[i];
        p[i] = (d > 0.f) ? rsqrtf(d) : 0.f;
    }
}

// zero-pad W (Fin x Fout) into 16x16 block
__global__ void pad_weights(const float* __restrict__ W, int Fin, int Fout,
                            float* __restrict__ Wp) {
    int t = threadIdx.x;                       // 256 threads
    int k = t >> 4, n = t & 15;
    Wp[t] = (k < Fin && n < Fout) ? W[k * Fout + n] : 0.f;
}

// zero-pad x (N x Fin) into (N x 16)
__global__ void pad_x(const float* __restrict__ x, float* __restrict__ xp,
                      int Fin, int N) {
    long long t = (long long)blockIdx.x * blockDim.x + threadIdx.x;
    if (t >= (long long)N * 16) return;
    int i = (int)(t >> 4), j = (int)(t & 15);
    xp[t] = (j < Fin) ? x[(long long)i * Fin + j] : 0.f;
}

// int64 edge list -> int32 src/dst
__global__ void cvt_edges(const long long* __restrict__ ei, int* __restrict__ s32,
                          int* __restrict__ d32, long long E) {
    long long e = (long long)blockIdx.x * blockDim.x + threadIdx.x;
    if (e >= E) return;
    s32[e] = (int)ei[e];
    d32[e] = (int)ei[E + e];
}

__global__ void deg_accum32(const int* __restrict__ dst, float* __restrict__ deg,
                            long long E) {
    long long e = (long long)blockIdx.x * blockDim.x + threadIdx.x;
    if (e >= E) return;
    __hip_atomic_fetch_add(&deg[dst[e]], 1.0f, __ATOMIC_RELAXED,
                           __HIP_MEMORY_SCOPE_AGENT);
}

__global__ void deg_accum64(const long long* __restrict__ dst, float* __restrict__ deg,
                            long long E) {
    long long e = (long long)blockIdx.x * blockDim.x + threadIdx.x;
    if (e >= E) return;
    __hip_atomic_fetch_add(&deg[(int)dst[e]], 1.0f, __ATOMIC_RELAXED,
                           __HIP_MEMORY_SCOPE_AGENT);
}

// norm[e] = dinv[src] * dinv[dst]
__global__ void edge_norm(const int* __restrict__ s32, const int* __restrict__ d32,
                          const float* __restrict__ dinv, float* __restrict__ nrm,
                          long long E) {
    long long e = (long long)blockIdx.x * blockDim.x + threadIdx.x;
    if (e >= E) return;
    nrm[e] = dinv[s32[e]] * dinv[d32[e]];
}

// ------------------------------------------------ WMMA feature transform
// Full 16-node tiles, inputs pre-padded to stride 16, weights pre-padded 16x16.
// One wave per tile; 4 chained V_WMMA_F32_16X16X4_F32, no per-lane guards.
__global__ void transform_wmma16(const float* __restrict__ in,
                                 const float* __restrict__ Wp,
                                 float* __restrict__ h, int numTiles) {
    int wave = threadIdx.x >> 5;
    int lane = threadIdx.x & 31;
    int tile = blockIdx.x * (blockDim.x >> 5) + wave;
    if (tile >= numTiles) return;          // wave-uniform; EXEC all-ones below

    int mr = lane & 15;                    // row within tile (A), col (B/store)
    int hi = lane >> 4;                    // half-wave K-split
    const float* rowp = in + ((long long)tile * 16 + mr) * 16;

    v8f c = {};
#pragma unroll
    for (int s = 0; s < 4; ++s) {
        int klo = s * 4 + hi * 2;
        v2f a = *(const v2f*)(rowp + klo);         // global_load_b64
        v2f b;
        b.x = Wp[klo * 16 + mr];
        b.y = Wp[klo * 16 + 16 + mr];
        c = __builtin_amdgcn_wmma_f32_16x16x4_f32(false, a, false, b, (short)0, c,
                                                  false, false);
    }

    // C/D layout: VGPR p -> row M = p (+8 for hi half), col = mr
    float* outp = h + (long long)tile * 256 + hi * 128 + mr;
#pragma unroll
    for (int p = 0; p < 8; ++p) outp[p * 16] = c[p];
}

// scalar tail for N % 16 remainder nodes (launched only if rem > 0)
__global__ void transform_tail(const float* __restrict__ in, const float* __restrict__ Wp,
                               float* __restrict__ h, int base, int rem) {
    int t = threadIdx.x;                   // 256 threads, one block
    int r = t >> 4, n = t & 15;
    if (r >= rem) return;
    long long row = base + r;
    float s = 0.f;
#pragma unroll
    for (int k = 0; k < 16; ++k) s += in[row * 16 + k] * Wp[k * 16 + n];
    h[row * 16 + n] = s;
}

// --------------------------------------------- edge gather + scatter-add
// 16 lanes per edge; h gather = coalesced 64B, scatter = global_atomic_add_f32
// resolved in the 192MB L2 (agg is fully L2-resident).
__global__ void edge_scatter32(const int* __restrict__ src, const int* __restrict__ dst,
                               const float* __restrict__ nrm,
                               const float* __restrict__ h,
                               float* __restrict__ agg, long long E) {
    long long t = (long long)blockIdx.x * blockDim.x + threadIdx.x;
    long long e = t >> 4;
    if (e >= E) return;
    int j = (int)(t & 15);
    int s = src[e];
    int d = dst[e];
    float val = h[(long long)s * 16 + j] * nrm[e];
    __hip_atomic_fetch_add(&agg[(long long)d * 16 + j], val, __ATOMIC_RELAXED,
                           __HIP_MEMORY_SCOPE_AGENT);
}

__global__ void edge_scatter64(const long long* __restrict__ src,
                               const long long* __restrict__ dst,
                               const float* __restrict__ dinv,
                               const float* __restrict__ h,
                               float* __restrict__ agg, long long E) {
    long long t = (long long)blockIdx.x * blockDim.x + threadIdx.x;
    long long e = t >> 4;
    if (e >= E) return;
    int j = (int)(t & 15);
    int s = (int)src[e];
    int d = (int)dst[e];
    float w = dinv[s] * dinv[d];
    float val = h[(long long)s * 16 + j] * w;
    __hip_atomic_fetch_add(&agg[(long long)d * 16 + j], val, __ATOMIC_RELAXED,
                           __HIP_MEMORY_SCOPE_AGENT);
}

// ------------------------------- self-loop + bias + ReLU (in place in agg)
__global__ void finalize_relu(float* __restrict__ agg, const float* __restrict__ h,
                              const float* __restrict__ dinv,
                              const float* __restrict__ bias, int Fout, int N) {
    long long t = (long long)blockIdx.x * blockDim.x + threadIdx.x;
    if (t >= (long long)N * 16) return;
    int i = (int)(t >> 4);
    int j = (int)(t & 15);
    float v = 0.f;
    if (j < Fout) {
        float dw = dinv[i];
        v = agg[t] + h[t] * dw * dw + bias[j];
        v = v > 0.f ? v : 0.f;
    }
    agg[t] = v;
}

// ---------------------- last layer: self-loop + bias + log_softmax(6)
__global__ void finalize_logsoftmax(const float* __restrict__ agg,
                                    const float* __restrict__ h,
                                    const float* __restrict__ dinv,
                                    const float* __restrict__ bias,
                                    float* __restrict__ out, int N) {
    int i = blockIdx.x * blockDim.x + threadIdx.x;
    if (i >= N) return;
    float dw = dinv[i];
    float sw = dw * dw;
    float v[6];
    float m = -3.0e38f;
#pragma unroll
    for (int j = 0; j < 6; ++j) {
        v[j] = agg[(long long)i * 16 + j] + h[(long long)i * 16 + j] * sw + bias[j];
        m = fmaxf(m, v[j]);
    }
    float sum = 0.f;
#pragma unroll
    for (int j = 0; j < 6; ++j) sum += expf(v[j] - m);
    float lse = logf(sum);
#pragma unroll
    for (int j = 0; j < 6; ++j) out[(long long)i * 6 + j] = v[j] - m - lse;
}

// ---------------------------------------------------------------- driver
extern "C" void kernel_launch(void* const* d_in, const int* in_sizes, int n_in,
                              void* d_out, int out_size, void* d_ws, size_t ws_size,
                              hipStream_t stream) {
    const float*     x  = (const float*)d_in[0];
    const long long* ei = (const long long*)d_in[1];   // int64 [2, E]
    const float* Wl[3]  = {(const float*)d_in[2], (const float*)d_in[4],
                           (const float*)d_in[6]};
    const float* bl[3]  = {(const float*)d_in[3], (const float*)d_in[5],
                           (const float*)d_in[7]};
    const int Fin[3]  = {6, 16, 16};
    const int Fout[3] = {16, 16, 6};
    float* out = (float*)d_out;

    const int       N = in_sizes[0] / 6;             // 100000
    const long long E = (long long)in_sizes[1] / 2;  // 3200000
    const long long* src64 = ei;
    const long long* dst64 = ei + E;

    // workspace: dinv[N] | hbuf[16N] | agg[16N] | Wp[256] | (src32,dst32:int[E], nrm[E])
    float* dinv = (float*)d_ws;
    float* hbuf = dinv + N;
    float* agg  = hbuf + (long long)N * 16;
    float* Wp   = agg  + (long long)N * 16;
    int*   s32  = (int*)(Wp + 256);
    int*   d32  = s32 + E;
    float* nrm  = (float*)(d32 + E);
    const size_t needBig = (size_t)((long long)33 * N + 256 + 3 * E) * 4;
    const bool big = ws_size >= needBig;             // ws_size fixed -> deterministic

    const long long NF  = (long long)N * 16;
    const int fullTiles = N / 16;
    const int rem       = N % 16;
    const int wavesPerBlk = TPB / 32;
    dim3 gTrans((fullTiles + wavesPerBlk - 1) / wavesPerBlk);
    dim3 gNode((N + TPB - 1) / TPB);
    dim3 gNF((int)((NF + TPB - 1) / TPB));
    dim3 gEdge((int)((E + TPB - 1) / TPB));
    dim3 gScat((int)((E * 16 + TPB - 1) / TPB));

    // ---- symmetric normalization: dinv = rsqrt(1 + in-degree)
    fill_f32<<<gNode, TPB, 0, stream>>>(dinv, 1.0f, N);
    if (big) {
        cvt_edges<<<gEdge, TPB, 0, stream>>>(ei, s32, d32, E);
        deg_accum32<<<gEdge, TPB, 0, stream>>>(d32, dinv, E);
    } else {
        deg_accum64<<<gEdge, TPB, 0, stream>>>(dst64, dinv, E);
    }
    rsqrt_inplace<<<gNode, TPB, 0, stream>>>(dinv, N);
    if (big) edge_norm<<<gEdge, TPB, 0, stream>>>(s32, d32, dinv, nrm, E);

    // ---- layer 1 input: zero-pad x (N x 6) -> agg (N x 16)
    pad_x<<<gNF, TPB, 0, stream>>>(x, agg, 6, N);

    for (int l = 0; l < 3; ++l) {
        // transform: hbuf = agg_padded @ Wp   (WMMA, full f32)
        pad_weights<<<1, 256, 0, stream>>>(Wl[l], Fin[l], Fout[l], Wp);
        if (fullTiles)
            transform_wmma16<<<gTrans, TPB, 0, stream>>>(agg, Wp, hbuf, fullTiles);
        if (rem)
            transform_tail<<<1, 256, 0, stream>>>(agg, Wp, hbuf, fullTiles * 16, rem);

        // aggregate: agg = scatter_add(norm * hbuf[src] -> dst)
        fill_f32<<<gNF, TPB, 0, stream>>>(agg, 0.0f, NF);
        if (big)
            edge_scatter32<<<gScat, TPB, 0, stream>>>(s32, d32, nrm, hbuf, agg, E);
        else
            edge_scatter64<<<gScat, TPB, 0, stream>>>(src64, dst64, dinv, hbuf, agg, E);

        // epilogue
        if (l < 2)
            finalize_relu<<<gNF, TPB, 0, stream>>>(agg, hbuf, dinv, bl[l], Fout[l], N);
        else
            finalize_logsoftmax<<<gNode, TPB, 0, stream>>>(agg, hbuf, dinv, bl[l], out, N);
    }
}